// GeoEEGSNN_53695681135180
// MI455X (gfx1250) — compile-verified
//
#include <hip/hip_runtime.h>
#include <math.h>

typedef __attribute__((ext_vector_type(2))) float v2f;
typedef __attribute__((ext_vector_type(8))) float v8f;

#define NB   64
#define NC   64
#define NT   1024
#define NT4  256

// ---------------------------------------------------------------------------
// Kernel 1: curv/tang 1x1-conv (channel GEMM) + pool4 + transpose, via
// V_WMMA_F32_16X16X4_F32. One wave per 16x16 output tile.
//   out[o,t] = sum_c W[o,c] * xb[b,ft,c,t],  ft = 1 (curv) or 2 (tang)
// Then pooled over t groups of 4 into OUT[t4][b][o]  ([T/4, B, 32]).
// ---------------------------------------------------------------------------
__global__ __launch_bounds__(32)
void geoeeg_map_pool_wmma(const float* __restrict__ xb,
                          const float* __restrict__ curvW,
                          const float* __restrict__ tangW,
                          float* __restrict__ curv_pooled,
                          float* __restrict__ tang_pooled) {
  const int lane  = threadIdx.x;
  const int tTile = blockIdx.x * 16;         // 64 tiles over T
  const int oTile = blockIdx.y * 16;         // 2 tiles over 32 outputs
  const int b     = blockIdx.z >> 1;
  const int which = blockIdx.z & 1;          // 0 = curv (ft=1), 1 = tang (ft=2)
  const float* W  = which ? tangW : curvW;
  const float* X  = xb + (size_t)(b * 3 + 1 + which) * NC * NT;
  float* OUT      = which ? tang_pooled : curv_pooled;

  const int halfSel = lane >> 4;             // 0: K=k0..k0+1 ; 1: K=k0+2..k0+3
  const int l15     = lane & 15;

  v8f acc = {};
  for (int k0 = 0; k0 < NC; k0 += 4) {
    const int kk = k0 + halfSel * 2;
    v2f a, bf;
    // A (16x4, MxK): lane l<16 -> M=l, K=kk,kk+1 in v0,v1; lanes 16-31 -> K=kk+2..
    a.x  = W[(oTile + l15) * NC + kk];
    a.y  = W[(oTile + l15) * NC + kk + 1];
    // B (4x16, KxN): mirrored K split, N striped over lanes
    bf.x = X[(size_t)kk * NT + tTile + l15];
    bf.y = X[(size_t)(kk + 1) * NT + tTile + l15];
    acc = __builtin_amdgcn_wmma_f32_16x16x4_f32(false, a, false, bf,
                                                (short)0, acc, false, false);
  }

  // C/D layout: VGPR p, lanes 0-15 -> M=p, lanes 16-31 -> M=p+8; N = l15.
  // pool4 over t: sum lanes within xor-groups {1,2} (stays inside half-wave).
#pragma unroll
  for (int p = 0; p < 8; ++p) {
    float v = acc[p];
    v += __shfl_xor(v, 1, 32);
    v += __shfl_xor(v, 2, 32);
    if ((l15 & 3) == 0) {
      const int t4 = (tTile + l15) >> 2;
      const int o  = oTile + p + 8 * halfSel;
      OUT[((size_t)t4 * NB + b) * 32 + o] = v * 0.25f;
    }
  }
}

// ---------------------------------------------------------------------------
// Kernel 2: population encoder + grouped spatial contraction.
//   s[b, g*2+d, t] = sum_c exp(-(xb[b,ft,c,t]-mu[k])^2 * inv2s[k]) * sW[g,d,c]
// with g = ft*4 + k. One thread per (b,t).
// ---------------------------------------------------------------------------
__global__ __launch_bounds__(256)
void geoeeg_enc_spatial(const float* __restrict__ xb,
                        const float* __restrict__ mu,
                        const float* __restrict__ sigma,
                        const float* __restrict__ spatialW,
                        float* __restrict__ s_raw) {
  __shared__ float sW[12 * 2 * 64];
  __shared__ float sMu[4], sInv[4];
  const int tid = threadIdx.x;
  for (int i = tid; i < 12 * 2 * 64; i += 256) sW[i] = spatialW[i];
  if (tid < 4) {
    sMu[tid] = mu[tid];
    const float s = sigma[tid];
    sInv[tid] = 1.0f / (2.0f * s * s + 1e-6f);
  }
  __syncthreads();

  const int flat = blockIdx.x * 256 + tid;   // 256 blocks * 256 = B*T threads
  const int b = flat >> 10;
  const int t = flat & 1023;

  float acc[24];
#pragma unroll
  for (int i = 0; i < 24; ++i) acc[i] = 0.0f;

  for (int ft = 0; ft < 3; ++ft) {
    const float* xp = xb + (size_t)(b * 3 + ft) * NC * NT + t;
    for (int c = 0; c < NC; ++c) {
      const float x = xp[(size_t)c * NT];
      if (c + 8 < NC) __builtin_prefetch(xp + (size_t)(c + 8) * NT, 0, 1);
#pragma unroll
      for (int k = 0; k < 4; ++k) {
        const float d = x - sMu[k];
        const float e = __expf(-d * d * sInv[k]);
        const int g = ft * 4 + k;
        acc[g * 2 + 0] += e * sW[(g * 2 + 0) * 64 + c];
        acc[g * 2 + 1] += e * sW[(g * 2 + 1) * 64 + c];
      }
    }
  }
  for (int ch = 0; ch < 24; ++ch)
    s_raw[(size_t)(b * 24 + ch) * NT + t] = acc[ch];
}

// ---------------------------------------------------------------------------
// Kernel 3/6: per-channel sum & sum-of-squares over (B,T). One block per
// channel, fixed-order tree reduce => deterministic.
// ---------------------------------------------------------------------------
__global__ __launch_bounds__(256)
void geoeeg_chan_stats(const float* __restrict__ buf, int nch,
                       float* __restrict__ sums, float* __restrict__ sumsq) {
  __shared__ float s1[256], s2[256];
  const int ch = blockIdx.x;
  const int tid = threadIdx.x;
  float a = 0.0f, q = 0.0f;
  for (int idx = tid; idx < NB * NT; idx += 256) {
    const int bb = idx >> 10, t = idx & 1023;
    const float v = buf[(size_t)(bb * nch + ch) * NT + t];
    a += v;
    q += v * v;
  }
  s1[tid] = a; s2[tid] = q;
  __syncthreads();
  for (int off = 128; off > 0; off >>= 1) {
    if (tid < off) { s1[tid] += s1[tid + off]; s2[tid] += s2[tid + off]; }
    __syncthreads();
  }
  if (tid == 0) { sums[ch] = s1[0]; sumsq[ch] = s2[0]; }
}

// ---------------------------------------------------------------------------
// Kernel 4: BN(24ch) + ELU + depthwise conv1d k=15 pad 7 (correlation).
// One block per (b,ch) row; BN+ELU applied while staging the row in LDS.
// ---------------------------------------------------------------------------
__global__ __launch_bounds__(256)
void geoeeg_bn2_elu_dwconv(const float* __restrict__ s_raw,
                           const float* __restrict__ sum24,
                           const float* __restrict__ sq24,
                           const float* __restrict__ bn2_g,
                           const float* __restrict__ bn2_b,
                           const float* __restrict__ dwW,
                           float* __restrict__ dwbuf) {
  __shared__ float buf[NT + 14];
  __shared__ float w[15];
  const int ch = blockIdx.x % 24;
  const int b  = blockIdx.x / 24;
  const int tid = threadIdx.x;

  const float invN  = 1.0f / (float)(NB * NT);
  const float mean  = sum24[ch] * invN;
  const float var   = sq24[ch] * invN - mean * mean;   // biased, matches jnp.var
  const float scale = rsqrtf(var + 1e-5f) * bn2_g[ch];
  const float shift = bn2_b[ch] - mean * scale;
  if (tid < 15) w[tid] = dwW[ch * 15 + tid];

  const float* row = s_raw + (size_t)(b * 24 + ch) * NT;
  for (int i = tid; i < NT + 14; i += 256) {
    const int pos = i - 7;
    float v = 0.0f;
    if (pos >= 0 && pos < NT) {
      const float x = row[pos] * scale + shift;
      v = x > 0.0f ? x : expm1f(x);                    // ELU, pad zeros outside
    }
    buf[i] = v;
  }
  __syncthreads();

  float* orow = dwbuf + (size_t)(b * 24 + ch) * NT;
#pragma unroll
  for (int j = 0; j < 4; ++j) {
    const int t = tid * 4 + j;
    float o = 0.0f;
#pragma unroll
    for (int i = 0; i < 15; ++i) o += w[i] * buf[t + i];
    orow[t] = o;
  }
}

// ---------------------------------------------------------------------------
// Kernel 5: pointwise conv (32x24 GEMM over T) via V_WMMA_F32_16X16X4_F32.
//   pw[b,o,t] = sum_ch pwW[o,ch] * dw[b,ch,t];  K=24 as six k=4 steps.
// ---------------------------------------------------------------------------
__global__ __launch_bounds__(32)
void geoeeg_pointwise_wmma(const float* __restrict__ dwbuf,
                           const float* __restrict__ pwW,
                           float* __restrict__ pwbuf) {
  const int lane  = threadIdx.x;
  const int tTile = blockIdx.x * 16;
  const int oTile = blockIdx.y * 16;
  const int b     = blockIdx.z;
  const int halfSel = lane >> 4;
  const int l15     = lane & 15;

  v8f acc = {};
  for (int k0 = 0; k0 < 24; k0 += 4) {
    const int kk = k0 + halfSel * 2;
    v2f a, bf;
    a.x  = pwW[(oTile + l15) * 24 + kk];
    a.y  = pwW[(oTile + l15) * 24 + kk + 1];
    bf.x = dwbuf[(size_t)(b * 24 + kk) * NT + tTile + l15];
    bf.y = dwbuf[(size_t)(b * 24 + kk + 1) * NT + tTile + l15];
    acc = __builtin_amdgcn_wmma_f32_16x16x4_f32(false, a, false, bf,
                                                (short)0, acc, false, false);
  }
#pragma unroll
  for (int p = 0; p < 8; ++p) {
    const int o = oTile + p + 8 * halfSel;
    pwbuf[(size_t)(b * 32 + o) * NT + tTile + l15] = acc[p];
  }
}

// ---------------------------------------------------------------------------
// Kernel 7: BN(32ch) + ELU, then pool4 over t -> c_seq[t4][b][o].
// ---------------------------------------------------------------------------
__global__ __launch_bounds__(256)
void geoeeg_bn1_elu_pool(const float* __restrict__ pwbuf,
                         const float* __restrict__ sum32,
                         const float* __restrict__ sq32,
                         const float* __restrict__ g1,
                         const float* __restrict__ b1,
                         float* __restrict__ c_seq) {
  const int flat = blockIdx.x * 256 + threadIdx.x;  // 2048 blocks = B*32*T4
  const int o  = flat & 31;
  const int b  = (flat >> 5) & 63;
  const int t4 = flat >> 11;

  const float invN  = 1.0f / (float)(NB * NT);
  const float mean  = sum32[o] * invN;
  const float var   = sq32[o] * invN - mean * mean;
  const float scale = rsqrtf(var + 1e-5f) * g1[o];
  const float shift = b1[o] - mean * scale;

  const float* row = pwbuf + (size_t)(b * 32 + o) * NT + t4 * 4;
  float acc = 0.0f;
#pragma unroll
  for (int j = 0; j < 4; ++j) {
    const float x = row[j] * scale + shift;
    acc += x > 0.0f ? x : expm1f(x);                 // ELU before pooling
  }
  c_seq[((size_t)t4 * NB + b) * 32 + o] = acc * 0.25f;
}

// ---------------------------------------------------------------------------
// Kernel 8: SmoothGDLIF scan. Recurrence is elementwise per (b,o): one
// thread walks all 256 timesteps. Heaviside forward (eval path).
// ---------------------------------------------------------------------------
__global__ __launch_bounds__(256)
void geoeeg_lif_scan(const float* __restrict__ c_seq,
                     const float* __restrict__ curv_pooled,
                     const float* __restrict__ tang_pooled,
                     const float* __restrict__ tau_geo,
                     const float* __restrict__ beta_m,
                     const float* __restrict__ alpha,
                     const float* __restrict__ gamma_p,
                     float* __restrict__ fr) {
  const int idx = blockIdx.x * 256 + threadIdx.x;    // 8 blocks = 2048 = B*32
  const int o = idx & 31;
  const int b = idx >> 5;
  const float tg = 1.0f / (1.0f + __expf(-tau_geo[0]));
  const float bm = beta_m[0], al = alpha[0], gp = gamma_p[0];

  float m = 0.0f, cs = 0.0f, ts = 0.0f, sum = 0.0f;
  for (int t = 0; t < NT4; ++t) {
    const size_t base = ((size_t)t * NB + b) * 32 + o;
    const float x  = c_seq[base];
    const float cu = curv_pooled[base];
    const float ta = tang_pooled[base];
    cs = tg * cs + (1.0f - tg) * cu;
    ts = tg * ts + (1.0f - tg) * ta;
    const float vth  = fmaxf(0.3f + al * cs, 0.1f);
    const float sens = 1.0f / (1.0f + __expf(-(1.0f - gp * ts)));
    m = bm * m + x * sens;
    const float sp = (m - vth) > 0.0f ? 1.0f : 0.0f;
    m -= sp * vth;
    sum += sp;
  }
  fr[b * 32 + o] = sum * (1.0f / (float)NT4);
}

// ---------------------------------------------------------------------------
// Kernel 9: final FC  out[b,j] = sum_o fr[b,o]*fcW[j,o] + fcb[j]
// ---------------------------------------------------------------------------
__global__ __launch_bounds__(256)
void geoeeg_fc(const float* __restrict__ fr, const float* __restrict__ fcW,
               const float* __restrict__ fcb, float* __restrict__ out) {
  const int idx = threadIdx.x;                       // 256 = 64*4
  const int j = idx & 3;
  const int b = idx >> 2;
  float acc = fcb[j];
#pragma unroll
  for (int o = 0; o < 32; ++o) acc += fr[b * 32 + o] * fcW[j * 32 + o];
  out[b * 4 + j] = acc;
}

// ---------------------------------------------------------------------------
extern "C" void kernel_launch(void* const* d_in, const int* in_sizes, int n_in,
                              void* d_out, int out_size, void* d_ws, size_t ws_size,
                              hipStream_t stream) {
  const float* xb       = (const float*)d_in[0];
  const float* mu       = (const float*)d_in[1];
  const float* sigma    = (const float*)d_in[2];
  const float* curvW    = (const float*)d_in[3];
  const float* tangW    = (const float*)d_in[4];
  const float* spatialW = (const float*)d_in[5];
  const float* bn2_g    = (const float*)d_in[6];
  const float* bn2_b    = (const float*)d_in[7];
  const float* dwW      = (const float*)d_in[8];
  const float* pwW      = (const float*)d_in[9];
  const float* bn1_g    = (const float*)d_in[10];
  const float* bn1_b    = (const float*)d_in[11];
  const float* tau_geo  = (const float*)d_in[12];
  const float* beta_m   = (const float*)d_in[13];
  const float* alpha    = (const float*)d_in[14];
  const float* gamma_p  = (const float*)d_in[15];
  const float* fcW      = (const float*)d_in[16];
  const float* fcb      = (const float*)d_in[17];
  float* out = (float*)d_out;

  // Workspace carve-up (floats). Total ~6.82M floats (~27.3 MB).
  float* ws = (float*)d_ws;
  float* curv_pooled = ws;                 ws += (size_t)NT4 * NB * 32;   // 524288
  float* tang_pooled = ws;                 ws += (size_t)NT4 * NB * 32;   // 524288
  float* s_raw       = ws;                 ws += (size_t)NB * 24 * NT;    // 1572864
  float* dwbuf       = ws;                 ws += (size_t)NB * 24 * NT;    // 1572864
  float* pwbuf       = ws;                 ws += (size_t)NB * 32 * NT;    // 2097152
  float* c_seq       = ws;                 ws += (size_t)NT4 * NB * 32;   // 524288
  float* sum24       = ws;                 ws += 24;
  float* sq24        = ws;                 ws += 24;
  float* sum32       = ws;                 ws += 32;
  float* sq32        = ws;                 ws += 32;
  float* fr          = ws;                 ws += (size_t)NB * 32;

  // 1) curv/tang channel GEMMs + pool4 (WMMA f32)
  geoeeg_map_pool_wmma<<<dim3(NT / 16, 2, NB * 2), 32, 0, stream>>>(
      xb, curvW, tangW, curv_pooled, tang_pooled);

  // 2) encoder + spatial grouped contraction
  geoeeg_enc_spatial<<<(NB * NT) / 256, 256, 0, stream>>>(
      xb, mu, sigma, spatialW, s_raw);

  // 3) BN2 stats
  geoeeg_chan_stats<<<24, 256, 0, stream>>>(s_raw, 24, sum24, sq24);

  // 4) BN2 + ELU + depthwise conv
  geoeeg_bn2_elu_dwconv<<<NB * 24, 256, 0, stream>>>(
      s_raw, sum24, sq24, bn2_g, bn2_b, dwW, dwbuf);

  // 5) pointwise GEMM (WMMA f32)
  geoeeg_pointwise_wmma<<<dim3(NT / 16, 2, NB), 32, 0, stream>>>(
      dwbuf, pwW, pwbuf);

  // 6) BN1 stats
  geoeeg_chan_stats<<<32, 256, 0, stream>>>(pwbuf, 32, sum32, sq32);

  // 7) BN1 + ELU + pool4
  geoeeg_bn1_elu_pool<<<(NB * 32 * NT4) / 256, 256, 0, stream>>>(
      pwbuf, sum32, sq32, bn1_g, bn1_b, c_seq);

  // 8) LIF scan
  geoeeg_lif_scan<<<(NB * 32) / 256, 256, 0, stream>>>(
      c_seq, curv_pooled, tang_pooled, tau_geo, beta_m, alpha, gamma_p, fr);

  // 9) FC head
  geoeeg_fc<<<1, 256, 0, stream>>>(fr, fcW, fcb, out);
}